// SinkhornOTLoss_21414706938293
// MI455X (gfx1250) — compile-verified
//
#include <hip/hip_runtime.h>
#include <math.h>

typedef __attribute__((ext_vector_type(2))) float v2f;
typedef __attribute__((ext_vector_type(8))) float v8f;

#define NPTS 8192
#define DIMK 128
#define INV_EPS 20.0f      // 1/0.05
#define N_ITERS 30
#define NCH 4              // column chunks (grid.y)
#define COLS_PER_CH (NPTS / NCH)
#define LROW 132           // padded LDS row stride (floats): conflict-free b64
#define TILE_F (16 * LROW) // floats per LDS tile buffer

// ---------------------------------------------------------------------------
// init: row squared norms + zero log_u/log_v
// ---------------------------------------------------------------------------
__global__ void __launch_bounds__(256) sk_init(const float* __restrict__ zA,
                                               const float* __restrict__ zB,
                                               float* __restrict__ a2,
                                               float* __restrict__ b2,
                                               float* __restrict__ lu,
                                               float* __restrict__ lv) {
  int i = blockIdx.x * blockDim.x + threadIdx.x;
  if (i >= NPTS) return;
  const float4* pa = (const float4*)(zA + (size_t)i * DIMK);
  const float4* pb = (const float4*)(zB + (size_t)i * DIMK);
  float sa = 0.f, sb = 0.f;
#pragma unroll 8
  for (int k = 0; k < DIMK / 4; ++k) {
    float4 a = pa[k];
    float4 b = pb[k];
    sa += a.x * a.x + a.y * a.y + a.z * a.z + a.w * a.w;
    sb += b.x * b.x + b.y * b.y + b.z * b.z + b.w * b.w;
  }
  a2[i] = sa;
  b2[i] = sb;
  lu[i] = 0.f;
  lv[i] = 0.f;
}

// ---------------------------------------------------------------------------
// Partial logsumexp over one column chunk. Each wave owns 16 rows; the 16x128
// B tile is staged cooperatively into double-buffered LDS (one 8KB tile per
// WG instead of 8 redundant per-wave VMEM streams), WMMA f32 16x16x4 rebuilds
// each 16x16 Gram tile from registers(A) x LDS(B).
// ---------------------------------------------------------------------------
__global__ void __launch_bounds__(256) sk_lse_pass(const float* __restrict__ X,
                                                   const float* __restrict__ Y,
                                                   const float* __restrict__ x2,
                                                   const float* __restrict__ y2,
                                                   const float* __restrict__ addv,
                                                   float* __restrict__ pm,
                                                   float* __restrict__ ps) {
  __shared__ float lds[2 * TILE_F];

  const int tid = threadIdx.x;
  const int lane = tid & 31;
  const int wave = tid >> 5;
  const int lrow = lane & 15;
  const int half = lane >> 4;                  // 0: lanes 0-15, 1: lanes 16-31
  const int rowbase = blockIdx.x * 128 + wave * 16;
  const int colbeg = blockIdx.y * COLS_PER_CH;
  const int colend = colbeg + COLS_PER_CH;

  // Staging assignment: 512 float4 per tile, 2 per thread.
  const int f0 = tid, f1 = tid + 256;
  const int l0 = (f0 >> 5) * LROW + (f0 & 31) * 4;  // 16B-aligned (528=33*16)
  const int l1 = (f1 >> 5) * LROW + (f1 & 31) * 4;

  // Preload this wave's 16xK A strip into registers (ISA 7.12.2 layout:
  // lane l holds row l%16, K pair 2*(l/16) in each 2-VGPR group).
  v2f Areg[DIMK / 4];
  const float* Aptr = X + (size_t)(rowbase + lrow) * DIMK + 2 * half;
#pragma unroll
  for (int ks = 0; ks < DIMK / 4; ++ks)
    Areg[ks] = *(const v2f*)(Aptr + 4 * ks);

  float a2r[8];
#pragma unroll
  for (int r = 0; r < 8; ++r) a2r[r] = x2[rowbase + r + 8 * half];

  float mrun[8], srun[8];
#pragma unroll
  for (int r = 0; r < 8; ++r) { mrun[r] = -INFINITY; srun[r] = 0.f; }

  // Prologue: stage first tile into buffer 0.
  {
    const float* g = Y + (size_t)colbeg * DIMK;
    *(float4*)&lds[l0] = *(const float4*)(g + 4 * f0);
    *(float4*)&lds[l1] = *(const float4*)(g + 4 * f1);
  }
  __syncthreads();

  int p = 0;
  for (int jb = colbeg; jb < colend; jb += 16) {
    // Issue next tile's global loads early (latency hidden by compute).
    const bool hasnext = (jb + 16 < colend);
    float4 n0, n1;
    if (hasnext) {
      const float* g = Y + (size_t)(jb + 16) * DIMK;
      n0 = *(const float4*)(g + 4 * f0);
      n1 = *(const float4*)(g + 4 * f1);
    }

    // Compute from LDS buffer p (conflict-free: bank = 4*lrow + 2*half).
    const float* Lb = &lds[p * TILE_F + lrow * LROW + 2 * half];
    v8f acc = {};
#pragma unroll
    for (int ks = 0; ks < DIMK / 4; ++ks) {
      v2f b = *(const v2f*)(Lb + 4 * ks);
      acc = __builtin_amdgcn_wmma_f32_16x16x4_f32(
          /*neg_a=*/false, Areg[ks], /*neg_b=*/false, b,
          /*c_mod=*/(short)0, acc, /*reuse_a=*/false, /*reuse_b=*/false);
    }

    const int col = jb + lrow;                 // D layout: column = lane%16
    const float addc = addv[col];
    const float b2c = y2[col];
#pragma unroll
    for (int r = 0; r < 8; ++r) {              // D row = r + 8*half
      float d = fmaf(acc[r], -2.0f, a2r[r] + b2c);
      float c = fmaxf(d, 0.0f);
      float t = fmaf(c, -INV_EPS, addc);       // M + add
      // one-exp online logsumexp update (init m=-inf -> e=0, s=1: correct)
      float e = expf(-fabsf(t - mrun[r]));
      srun[r] = (t > mrun[r]) ? fmaf(srun[r], e, 1.0f) : (srun[r] + e);
      mrun[r] = fmaxf(mrun[r], t);
    }

    // Stage next tile into the other buffer (its readers finished last iter).
    if (hasnext) {
      float* d = &lds[(1 - p) * TILE_F];
      *(float4*)&d[l0] = n0;
      *(float4*)&d[l1] = n1;
    }
    __syncthreads();
    p ^= 1;
  }

  // Reduce (max,sum) across the 16 lanes of each half (masks stay in-half).
#pragma unroll
  for (int r = 0; r < 8; ++r) {
    float m = mrun[r], s = srun[r];
#pragma unroll
    for (int off = 1; off < 16; off <<= 1) {
      float mo = __shfl_xor(m, off, 32);
      float so = __shfl_xor(s, off, 32);
      float e = expf(-fabsf(m - mo));
      s = (m > mo) ? fmaf(so, e, s) : fmaf(s, e, so);
      m = fmaxf(m, mo);
    }
    if (lrow == 0) {
      int row = rowbase + r + 8 * half;
      pm[row * NCH + blockIdx.y] = m;
      ps[row * NCH + blockIdx.y] = s;
    }
  }
}

// ---------------------------------------------------------------------------
// Combine per-chunk (m,s) states: out[i] = logconst - (m_tot + log(s_tot))
// ---------------------------------------------------------------------------
__global__ void __launch_bounds__(256) sk_combine(const float* __restrict__ pm,
                                                  const float* __restrict__ ps,
                                                  float* __restrict__ outv,
                                                  float logconst) {
  int i = blockIdx.x * blockDim.x + threadIdx.x;
  if (i >= NPTS) return;
  float m = pm[i * NCH], s = ps[i * NCH];
#pragma unroll
  for (int c = 1; c < NCH; ++c) {
    float mo = pm[i * NCH + c], so = ps[i * NCH + c];
    float e = expf(-fabsf(m - mo));
    s = (m > mo) ? fmaf(so, e, s) : fmaf(s, e, so);
    m = fmaxf(m, mo);
  }
  outv[i] = logconst - (m + logf(s));
}

// ---------------------------------------------------------------------------
// Final loss: partials[block] = sum over block's tile range of exp(M+lu+lv)*C
// ---------------------------------------------------------------------------
__global__ void __launch_bounds__(256) sk_loss_pass(const float* __restrict__ X,
                                                    const float* __restrict__ Y,
                                                    const float* __restrict__ x2,
                                                    const float* __restrict__ y2,
                                                    const float* __restrict__ lu,
                                                    const float* __restrict__ lv,
                                                    float* __restrict__ partials) {
  __shared__ float lds[2 * TILE_F];
  __shared__ float wsum[8];

  const int tid = threadIdx.x;
  const int lane = tid & 31;
  const int wave = tid >> 5;
  const int lrow = lane & 15;
  const int half = lane >> 4;
  const int rowbase = blockIdx.x * 128 + wave * 16;
  const int colbeg = blockIdx.y * COLS_PER_CH;
  const int colend = colbeg + COLS_PER_CH;

  const int f0 = tid, f1 = tid + 256;
  const int l0 = (f0 >> 5) * LROW + (f0 & 31) * 4;
  const int l1 = (f1 >> 5) * LROW + (f1 & 31) * 4;

  v2f Areg[DIMK / 4];
  const float* Aptr = X + (size_t)(rowbase + lrow) * DIMK + 2 * half;
#pragma unroll
  for (int ks = 0; ks < DIMK / 4; ++ks)
    Areg[ks] = *(const v2f*)(Aptr + 4 * ks);

  float a2r[8], lur[8];
#pragma unroll
  for (int r = 0; r < 8; ++r) {
    int row = rowbase + r + 8 * half;
    a2r[r] = x2[row];
    lur[r] = lu[row];
  }

  {
    const float* g = Y + (size_t)colbeg * DIMK;
    *(float4*)&lds[l0] = *(const float4*)(g + 4 * f0);
    *(float4*)&lds[l1] = *(const float4*)(g + 4 * f1);
  }
  __syncthreads();

  float sum = 0.f;
  int p = 0;
  for (int jb = colbeg; jb < colend; jb += 16) {
    const bool hasnext = (jb + 16 < colend);
    float4 n0, n1;
    if (hasnext) {
      const float* g = Y + (size_t)(jb + 16) * DIMK;
      n0 = *(const float4*)(g + 4 * f0);
      n1 = *(const float4*)(g + 4 * f1);
    }

    const float* Lb = &lds[p * TILE_F + lrow * LROW + 2 * half];
    v8f acc = {};
#pragma unroll
    for (int ks = 0; ks < DIMK / 4; ++ks) {
      v2f b = *(const v2f*)(Lb + 4 * ks);
      acc = __builtin_amdgcn_wmma_f32_16x16x4_f32(false, Areg[ks], false, b,
                                                  (short)0, acc, false, false);
    }

    const int col = jb + lrow;
    const float lvc = lv[col];
    const float b2c = y2[col];
#pragma unroll
    for (int r = 0; r < 8; ++r) {
      float d = fmaf(acc[r], -2.0f, a2r[r] + b2c);
      float c = fmaxf(d, 0.0f);
      float t = fmaf(c, -INV_EPS, lur[r] + lvc);  // log T_ij
      sum += expf(t) * c;
    }

    if (hasnext) {
      float* d = &lds[(1 - p) * TILE_F];
      *(float4*)&d[l0] = n0;
      *(float4*)&d[l1] = n1;
    }
    __syncthreads();
    p ^= 1;
  }

  // Wave reduce (all 32 lanes hold partials of the block sum).
#pragma unroll
  for (int off = 1; off < 32; off <<= 1) sum += __shfl_xor(sum, off, 32);

  if (lane == 0) wsum[wave] = sum;
  __syncthreads();
  if (tid == 0) {
    float t = 0.f;
#pragma unroll
    for (int w = 0; w < 8; ++w) t += wsum[w];
    partials[blockIdx.y * 64 + blockIdx.x] = t;
  }
}

__global__ void sk_finalize(const float* __restrict__ partials,
                            float* __restrict__ out) {
  float t = 0.f;
  for (int i = 0; i < 64 * NCH; ++i) t += partials[i];
  out[0] = t;
}

// ---------------------------------------------------------------------------
extern "C" void kernel_launch(void* const* d_in, const int* in_sizes, int n_in,
                              void* d_out, int out_size, void* d_ws, size_t ws_size,
                              hipStream_t stream) {
  const float* zA = (const float*)d_in[0];
  const float* zB = (const float*)d_in[1];
  float* ws = (float*)d_ws;
  float* a2 = ws;                       // 8192
  float* b2 = ws + NPTS;                // 8192
  float* lu = ws + 2 * NPTS;            // 8192
  float* lv = ws + 3 * NPTS;            // 8192
  float* pm = ws + 4 * NPTS;            // 8192*NCH
  float* ps = pm + NPTS * NCH;          // 8192*NCH
  float* partials = ps + NPTS * NCH;    // 64*NCH

  const float logab = -logf((float)NPTS);  // log_a == log_b (uniform)
  const dim3 grid2d(64, NCH);

  sk_init<<<NPTS / 256, 256, 0, stream>>>(zA, zB, a2, b2, lu, lv);

  for (int it = 0; it < N_ITERS; ++it) {
    // log_u = log_a - lse_j(M + log_v)
    sk_lse_pass<<<grid2d, 256, 0, stream>>>(zA, zB, a2, b2, lv, pm, ps);
    sk_combine<<<NPTS / 256, 256, 0, stream>>>(pm, ps, lu, logab);
    // log_v = log_b - lse_i(M + log_u)  (same kernel, A/B roles swapped)
    sk_lse_pass<<<grid2d, 256, 0, stream>>>(zB, zA, b2, a2, lu, pm, ps);
    sk_combine<<<NPTS / 256, 256, 0, stream>>>(pm, ps, lv, logab);
  }

  sk_loss_pass<<<grid2d, 256, 0, stream>>>(zA, zB, a2, b2, lu, lv, partials);
  sk_finalize<<<1, 1, 0, stream>>>(partials, (float*)d_out);
}